// LSTM_62517543960620
// MI455X (gfx1250) — compile-verified
//
#include <hip/hip_runtime.h>
#include <math.h>

typedef __attribute__((ext_vector_type(2))) float v2f;
typedef __attribute__((ext_vector_type(8))) float v8f;

#define T_STEPS 131072
#define HID 10
#define NG 40          // 4*H gates
#define PF 8           // software-pipeline depth for the xw0 / x stream

// ---------------------------------------------------------------------------
// Phase 1 (parallel, WMMA): xw0[t][g] = sum_k x[t,k]*Wih0[g,k] + bih0[g]+bhh0[g]
// One wave handles one 16(time) x 40(gate) tile: 3 x V_WMMA_F32_16X16X4_F32.
// ---------------------------------------------------------------------------
__global__ __launch_bounds__(256) void lstm_xw0_wmma(
    const float* __restrict__ x, const float* __restrict__ Wih0,
    const float* __restrict__ bih0, const float* __restrict__ bhh0,
    float* __restrict__ xw0)
{
  const int lane = threadIdx.x & 31;
  const int wave = threadIdx.x >> 5;
  const int tile = blockIdx.x * 8 + wave;      // 16-timestep tile index
  const int t0   = tile * 16;
  const int m    = lane & 15;
  const int koff = (lane < 16) ? 0 : 2;

  // A: 16x4 f32, lane L<16 -> (K=0,K=1) of row M=L ; lane L>=16 -> (K=2,K=3)
  v2f a;
  a.x = x[(t0 + m) * 4 + koff + 0];
  a.y = x[(t0 + m) * 4 + koff + 1];

#pragma unroll
  for (int n = 0; n < 3; ++n) {
    const int g  = n * 16 + m;
    const int gc = (g < NG) ? g : (NG - 1);    // clamp so loads stay uniform (EXEC all-1s)
    // B: 4x16 f32, lane L<16 -> rows K=0,1 at column N=L ; L>=16 -> rows K=2,3
    v2f b;
    b.x = Wih0[gc * 4 + koff + 0];
    b.y = Wih0[gc * 4 + koff + 1];
    float bv = bih0[gc] + bhh0[gc];
    if (g >= NG) { b.x = 0.0f; b.y = 0.0f; bv = 0.0f; }  // branchless select / rejoined EXEC

    // C depends only on N(gate): every row VGPR = bias[g]
    v8f c;
#pragma unroll
    for (int r = 0; r < 8; ++r) c[r] = bv;

    c = __builtin_amdgcn_wmma_f32_16x16x4_f32(
        /*neg_a=*/false, a, /*neg_b=*/false, b,
        /*c_mod=*/(short)0, c, /*reuse_a=*/false, /*reuse_b=*/false);

    if (g < NG) {
#pragma unroll
      for (int r = 0; r < 8; ++r) {
        const int M = (lane < 16) ? r : (r + 8);
        xw0[(t0 + M) * NG + g] = c[r];
      }
    }
  }
}

// ---------------------------------------------------------------------------
// Phase 2 (sequential, persistent single wave32)
// Lane ln owns gate ln (and gate ln+32 for ln<8). h0/h1 replicated per lane,
// c owned by lanes 0..9. xw0 (or x) stream is register-pipelined PF steps ahead.
// ---------------------------------------------------------------------------
__device__ __forceinline__ float gate_act(int type, float v) {
  // PyTorch gate order i,f,g,o -> type 2 (g) is tanh, rest sigmoid
  return (type == 2) ? tanhf(v) : (1.0f / (1.0f + __expf(-v)));
}

template <bool USE_WS>
__global__ __launch_bounds__(32) void lstm_seq(
    const float* __restrict__ xw0,
    const float* __restrict__ x,
    const float* __restrict__ h_out,
    const float* __restrict__ Wih0, const float* __restrict__ Whh0,
    const float* __restrict__ bih0, const float* __restrict__ bhh0,
    const float* __restrict__ Wih1, const float* __restrict__ Whh1,
    const float* __restrict__ bih1, const float* __restrict__ bhh1,
    const float* __restrict__ fc_w, const float* __restrict__ fc_b,
    float* __restrict__ out)
{
  const int ln = threadIdx.x;                  // 0..31
  const int g1 = ln;                           // gates 0..31
  const int g2 = (ln < 8) ? (ln + 32) : ln;    // gates 32..39 on lanes 0..7 (clamped)
  const int tA = g1 / HID;
  const int tB = g2 / HID;

  // per-lane weight rows in VGPRs
  float Wh0a[HID], Wh0b[HID], Wi1a[HID], Wi1b[HID], Wh1a[HID], Wh1b[HID];
#pragma unroll
  for (int k = 0; k < HID; ++k) {
    Wh0a[k] = Whh0[g1 * HID + k];  Wh0b[k] = Whh0[g2 * HID + k];
    Wi1a[k] = Wih1[g1 * HID + k];  Wi1b[k] = Wih1[g2 * HID + k];
    Wh1a[k] = Whh1[g1 * HID + k];  Wh1b[k] = Whh1[g2 * HID + k];
  }
  const float b1a = bih1[g1] + bhh1[g1];
  const float b1b = bih1[g2] + bhh1[g2];

  float Wi0a[4], Wi0b[4], bs0a = 0.0f, bs0b = 0.0f;
  if (!USE_WS) {
#pragma unroll
    for (int k = 0; k < 4; ++k) { Wi0a[k] = Wih0[g1 * 4 + k]; Wi0b[k] = Wih0[g2 * 4 + k]; }
    bs0a = bih0[g1] + bhh0[g1];
    bs0b = bih0[g2] + bhh0[g2];
  }

  float fcw[HID];
#pragma unroll
  for (int k = 0; k < HID; ++k) fcw[k] = fc_w[k];
  const float fcb = fc_b[0];

  float h0r[HID], h1r[HID];
#pragma unroll
  for (int k = 0; k < HID; ++k) { h0r[k] = h_out[k]; h1r[k] = h_out[HID + k]; }
  float c0 = 0.0f, c1 = 0.0f;                  // lane-owned (lanes < HID)

  __shared__ float sg[NG + 8];
  __shared__ float sh[16];

  // prime the prefetch pipeline
  float pfA[PF], pfB[PF];
  float4 pfX[PF];
#pragma unroll
  for (int p = 0; p < PF; ++p) {
    if (USE_WS) {
      pfA[p] = xw0[p * NG + g1];
      pfB[p] = xw0[p * NG + g2];
    } else {
      pfX[p] = *reinterpret_cast<const float4*>(x + p * 4);
    }
  }

  for (int t = 0; t < T_STEPS; t += PF) {
#pragma unroll
    for (int u = 0; u < PF; ++u) {
      const int tt = t + u;

      // ---- layer 0 gate pre-activation ----
      float preA, preB;
      if (USE_WS) {
        preA = pfA[u];
        preB = pfB[u];
      } else {
        const float4 xv = pfX[u];
        preA = bs0a + Wi0a[0]*xv.x + Wi0a[1]*xv.y + Wi0a[2]*xv.z + Wi0a[3]*xv.w;
        preB = bs0b + Wi0b[0]*xv.x + Wi0b[1]*xv.y + Wi0b[2]*xv.z + Wi0b[3]*xv.w;
      }
      // issue prefetch PF steps ahead (in-flight, consumed later)
      const int tp = tt + PF;
      if (tp < T_STEPS) {
        if (USE_WS) {
          pfA[u] = xw0[tp * NG + g1];
          pfB[u] = xw0[tp * NG + g2];
        } else {
          pfX[u] = *reinterpret_cast<const float4*>(x + tp * 4);
        }
      }

#pragma unroll
      for (int k = 0; k < HID; ++k) {
        preA = fmaf(Wh0a[k], h0r[k], preA);
        preB = fmaf(Wh0b[k], h0r[k], preB);
      }
      sg[g1] = gate_act(tA, preA);
      if (ln < 8) sg[g2] = gate_act(tB, preB);
      __syncthreads();

      if (ln < HID) {
        const float iv = sg[ln], fv = sg[HID + ln], gv = sg[2 * HID + ln], ov = sg[3 * HID + ln];
        c0 = fmaf(fv, c0, iv * gv);
        sh[ln] = ov * tanhf(c0);
      }
      __syncthreads();
#pragma unroll
      for (int k = 0; k < HID; ++k) h0r[k] = sh[k];

      // ---- layer 1 ----
      float p1A = b1a, p1B = b1b;
#pragma unroll
      for (int k = 0; k < HID; ++k) {
        p1A = fmaf(Wi1a[k], h0r[k], p1A);
        p1A = fmaf(Wh1a[k], h1r[k], p1A);
        p1B = fmaf(Wi1b[k], h0r[k], p1B);
        p1B = fmaf(Wh1b[k], h1r[k], p1B);
      }
      sg[g1] = gate_act(tA, p1A);
      if (ln < 8) sg[g2] = gate_act(tB, p1B);
      __syncthreads();

      if (ln < HID) {
        const float iv = sg[ln], fv = sg[HID + ln], gv = sg[2 * HID + ln], ov = sg[3 * HID + ln];
        c1 = fmaf(fv, c1, iv * gv);
        sh[ln] = ov * tanhf(c1);
      }
      __syncthreads();
#pragma unroll
      for (int k = 0; k < HID; ++k) h1r[k] = sh[k];

      // ---- FC head ----
      float acc = fcb;
#pragma unroll
      for (int k = 0; k < HID; ++k) acc = fmaf(fcw[k], h1r[k], acc);
      if (ln == 0) out[tt] = acc;
    }
  }

  // h_final = stack(hT0, hT1) appended after out[T]
  if (ln < HID) {
    out[T_STEPS + ln]       = h0r[ln];
    out[T_STEPS + HID + ln] = h1r[ln];
  }
}

// ---------------------------------------------------------------------------
extern "C" void kernel_launch(void* const* d_in, const int* in_sizes, int n_in,
                              void* d_out, int out_size, void* d_ws, size_t ws_size,
                              hipStream_t stream) {
  const float* x    = (const float*)d_in[0];
  const float* hout = (const float*)d_in[1];
  const float* Wih0 = (const float*)d_in[2];
  const float* Whh0 = (const float*)d_in[3];
  const float* bih0 = (const float*)d_in[4];
  const float* bhh0 = (const float*)d_in[5];
  const float* Wih1 = (const float*)d_in[6];
  const float* Whh1 = (const float*)d_in[7];
  const float* bih1 = (const float*)d_in[8];
  const float* bhh1 = (const float*)d_in[9];
  const float* fcw  = (const float*)d_in[10];
  const float* fcb  = (const float*)d_in[11];
  float* out = (float*)d_out;
  float* xw0 = (float*)d_ws;

  const bool use_ws = ws_size >= (size_t)T_STEPS * NG * sizeof(float);

  if (use_ws) {
    // 8192 tiles of 16 timesteps, 8 waves (tiles) per 256-thread block
    lstm_xw0_wmma<<<T_STEPS / 16 / 8, 256, 0, stream>>>(x, Wih0, bih0, bhh0, xw0);
    lstm_seq<true><<<1, 32, 0, stream>>>(xw0, x, hout, Wih0, Whh0, bih0, bhh0,
                                         Wih1, Whh1, bih1, bhh1, fcw, fcb, out);
  } else {
    lstm_seq<false><<<1, 32, 0, stream>>>(xw0, x, hout, Wih0, Whh0, bih0, bhh0,
                                          Wih1, Whh1, bih1, bhh1, fcw, fcb, out);
  }
}